// RoadLoss_1211180778005
// MI455X (gfx1250) — compile-verified
//
#include <hip/hip_runtime.h>
#include <hip/hip_bf16.h>
#include <math.h>

// ---------------------------------------------------------------------------
// RoadLoss on MI455X (gfx1250):
//   d2[n,cell] = (r^2+c^2)*1 + 1*(p0^2+p1^2) + r*(-2 p0) + c*(-2 p1)
// is a rank-4 bilinear form -> V_WMMA_F32_16X16X4_F32 computes a 16x16
// (points x cells) tile of exact-f32 squared distances per instruction.
// Mask bias is folded into the WMMA C-matrix (0 for included cells, +1e9 for
// excluded) so each tile is 2 WMMAs + 8 v_min3 ops.
//
// Data movement: each block stages its 16-row x 1024-col map strip (64 KB)
// into LDS with a single Tensor Data Mover transfer (tensor_load_to_lds,
// 6-arg clang-23 form), synchronized with s_wait_tensorcnt + workgroup
// barrier. Masks are then read from LDS (ds_load_b32), keeping the VMEM pipe
// free for prefetch of the next strip.
// ---------------------------------------------------------------------------

typedef __attribute__((ext_vector_type(2))) float v2f;
typedef __attribute__((ext_vector_type(8))) float v8f;
typedef __attribute__((ext_vector_type(4))) unsigned int v4u;
typedef __attribute__((ext_vector_type(8))) int v8i;
typedef __attribute__((ext_vector_type(4))) int v4i;

#define MAP_H 1024
#define MAP_W 1024
#define NPTS  128
#define BIGF  1.0e9f

#if defined(__has_builtin)
#if __has_builtin(__builtin_amdgcn_tensor_load_to_lds)
#define HAVE_TDM 1
#endif
#endif

__global__ void __launch_bounds__(256)
road_min_wmma_kernel(const float* __restrict__ hd_map,
                     const int*   __restrict__ pred,
                     unsigned int* __restrict__ min_nd_bits,
                     unsigned int* __restrict__ min_dr_bits)
{
    const int lane  = threadIdx.x & 31;
    const int wave  = threadIdx.x >> 5;    // 0..7 waves per block
    const int ptile = blockIdx.x;          // 0..7 : which 16-point tile
    const int strip = blockIdx.y;          // 0..63: which 16-row strip
    const int m     = lane & 15;
    const bool hi   = lane >= 16;

    // 64 KB LDS strip: 16 map rows x 1024 f32 cells
    __shared__ float tilebuf[16 * MAP_W];

#ifdef HAVE_TDM
    if (wave == 0) {
        // ---- Tensor DMA descriptor (D#) for a 2D 1024x16 f32 tile ----
        const unsigned long long ga =
            (unsigned long long)(const void*)hd_map +
            (unsigned long long)strip * 16ull * MAP_W * 4ull;  // tile start
        const unsigned int ldsoff = (unsigned int)(uintptr_t)(void*)tilebuf;

        v4u g0;
        g0.x = 1u;                                   // count=1, user mode
        g0.y = ldsoff;                               // lds_addr (bytes)
        g0.z = (unsigned int)ga;                     // global_addr[31:0]
        g0.w = ((unsigned int)(ga >> 32) & 0x01FFFFFFu)
             | (2u << 30);                           // addr[56:32] | type=2

        v8i g1;
        g1[0] = (int)(2u << 16);                     // data_size=2 (4 bytes)
        g1[1] = (int)((MAP_W & 0xFFFFu) << 16);      // tensor_dim0[15:0]
        g1[2] = (int)((MAP_W >> 16) | ((MAP_W & 0xFFFFu) << 16)); // td0 hi | td1 lo
        g1[3] = (int)((MAP_H >> 16) | ((MAP_W & 0xFFFFu) << 16)); // td1 hi | tile_dim0
        g1[4] = 16;                                  // tile_dim1=16 rows, tile_dim2=0
        g1[5] = MAP_W;                               // tensor_dim0_stride[31:0]
        g1[6] = 0;                                   // stride0 hi | stride1 lo
        g1[7] = 0;                                   // stride1 hi

        v4i g2 = {0, 0, 0, 0};                       // unused for 2D tile
        v4i g3 = {0, 0, 0, 0};
        v8i g4 = {0, 0, 0, 0, 0, 0, 0, 0};           // unused trailing group

        __builtin_amdgcn_tensor_load_to_lds(g0, g1, g2, g3, g4, 0);
        __builtin_amdgcn_s_wait_tensorcnt(0);
    }
    __syncthreads();
#else
    // Fallback: stage strip into LDS with plain vector loads
    for (int idx = threadIdx.x; idx < 16 * MAP_W; idx += 256)
        tilebuf[idx] = hd_map[(size_t)strip * 16 * MAP_W + idx];
    __syncthreads();
#endif

    // --- A matrix (16x4 f32): lanes 0-15 hold {K0,K1}, lanes 16-31 {K2,K3}
    const int   pidx = ptile * 16 + m;
    const float p0   = (float)pred[2 * pidx + 0];
    const float p1   = (float)pred[2 * pidx + 1];
    v2f a;
    if (!hi) { a.x = 1.0f;        a.y = p0 * p0 + p1 * p1; }  // K0, K1
    else     { a.x = -2.0f * p0;  a.y = -2.0f * p1;        }  // K2, K3

    v8f acc_nd, acc_dr;
#pragma unroll
    for (int i = 0; i < 8; ++i) { acc_nd[i] = __builtin_inff(); acc_dr[i] = __builtin_inff(); }

    const int rbase = wave * 2;                // 2 LDS rows per wave
    for (int rl = rbase; rl < rbase + 2; ++rl) {
        const int   r   = strip * 16 + rl;
        const float rr  = (float)r;
        const float rr2 = rr * rr;
        const float* __restrict__ mrow = tilebuf + rl * MAP_W;
        // pull the NEXT strip's rows toward L2 ahead of the next block's TDM
        __builtin_prefetch(hd_map + ((size_t)r + 16) * MAP_W + m * 64, 0, 0);

        for (int c0 = 0; c0 < MAP_W; c0 += 16) {
            const int   cell = c0 + m;
            const float cc   = (float)cell;
            const float mask = mrow[cell];     // ds_load_b32

            // --- B matrix (4x16 f32): lanes 0-15 rows {K0,K1}, 16-31 {K2,K3}
            v2f b;
            if (!hi) { b.x = rr2 + cc * cc; b.y = 1.0f; }  // K0, K1
            else     { b.x = rr;            b.y = cc;   }  // K2, K3

            // mask is exactly 0.0f or 1.0f -> arithmetic bias, no compares
            const float bias_dr = BIGF * mask;         // exclude non-road cells
            const float bias_nd = BIGF - bias_dr;      // exclude road cells
            v8f cnd, cdr;
#pragma unroll
            for (int i = 0; i < 8; ++i) { cnd[i] = bias_nd; cdr[i] = bias_dr; }

            v8f dnd = __builtin_amdgcn_wmma_f32_16x16x4_f32(
                false, a, false, b, (short)0, cnd, false, false);
            v8f ddr = __builtin_amdgcn_wmma_f32_16x16x4_f32(
                false, a, false, b, (short)0, cdr, false, false);

#pragma unroll
            for (int i = 0; i < 8; ++i) {
                acc_nd[i] = fminf(acc_nd[i], dnd[i]);
                acc_dr[i] = fminf(acc_dr[i], ddr[i]);
            }
        }
    }

    // --- cross-lane min within each 16-lane half.
    // D layout: VGPR v, lanes 0-15 -> point row v ; lanes 16-31 -> row v+8.
#pragma unroll
    for (int v = 0; v < 8; ++v) {
        float vn = acc_nd[v];
        float vd = acc_dr[v];
#pragma unroll
        for (int off = 8; off >= 1; off >>= 1) {
            vn = fminf(vn, __shfl_xor(vn, off, 32));
            vd = fminf(vd, __shfl_xor(vd, off, 32));
        }
        if (m == 0) {
            const int gp = ptile * 16 + v + (hi ? 8 : 0);
            // values are all >= 0 so uint bit-pattern ordering == float ordering
            atomicMin(&min_nd_bits[gp], __float_as_uint(vn));
            atomicMin(&min_dr_bits[gp], __float_as_uint(vd));
        }
    }
}

__global__ void road_init_kernel(unsigned int* __restrict__ nd,
                                 unsigned int* __restrict__ dr)
{
    const int i = threadIdx.x;
    if (i < NPTS) {
        nd[i] = 0x7F800000u;  // +inf
        dr[i] = 0x7F800000u;
    }
}

__global__ void __launch_bounds__(128)
road_finalize_kernel(const float* __restrict__ hd_map,
                     const int*   __restrict__ pred,
                     const unsigned int* __restrict__ ndb,
                     const unsigned int* __restrict__ drb,
                     float* __restrict__ out)
{
    __shared__ float sh[NPTS];
    const int n  = threadIdx.x;
    const int p0 = pred[2 * n + 0];
    const int p1 = pred[2 * n + 1];

    const bool outside_frame = (p0 < 0) | (p0 > MAP_H) | (p1 < 0) | (p1 > MAP_W);

    bool outside_road = false;
#pragma unroll
    for (int dr = -1; dr <= 0; ++dr) {
#pragma unroll
        for (int dc = -1; dc <= 0; ++dc) {
            const bool rv = (dr == -1) ? (p0 >= 1) : ((p0 >= 1) && (p0 < MAP_H));
            const bool cv = (dc == -1) ? (p1 >= 1) : ((p1 >= 1) && (p1 < MAP_W));
            int r = p0 + dr; r = r < 0 ? 0 : (r > MAP_H - 1 ? MAP_H - 1 : r);
            int c = p1 + dc; c = c < 0 ? 0 : (c > MAP_W - 1 ? MAP_W - 1 : c);
            const float val = hd_map[(size_t)r * MAP_W + c];
            outside_road = outside_road || (rv && cv && (val == 1.0f));
        }
    }

    const float min_nd = __uint_as_float(ndb[n]);
    const float min_dr = __uint_as_float(drb[n]);

    // exp(sqrt(min_dr)*ln2/K2) - 1 == exp2(sqrt(min_dr)/K2) - 1
    const float loss_off = exp2f(sqrtf(min_dr) * (1.0f / 40.0f)) - 1.0f;
    const float loss_in  = expf(-min_nd * (1.0f / 21.7f));

    sh[n] = outside_frame ? 0.0f : (outside_road ? loss_off : loss_in);
    __syncthreads();

#pragma unroll
    for (int s = 64; s >= 1; s >>= 1) {
        if (n < s) sh[n] += sh[n + s];
        __syncthreads();
    }
    if (n == 0) out[0] = sh[0] * (1.0f / (float)NPTS);
}

extern "C" void kernel_launch(void* const* d_in, const int* in_sizes, int n_in,
                              void* d_out, int out_size, void* d_ws, size_t ws_size,
                              hipStream_t stream)
{
    const float* hd_map = (const float*)d_in[0];   // (1024,1024) f32
    const int*   pred   = (const int*)d_in[1];     // (128,2) i32
    float*       out    = (float*)d_out;           // scalar mean

    unsigned int* ndb = (unsigned int*)d_ws;       // 128 uints
    unsigned int* drb = ndb + NPTS;                // 128 uints

    road_init_kernel<<<1, 128, 0, stream>>>(ndb, drb);

    dim3 grid(NPTS / 16, MAP_H / 16);              // (8 point-tiles, 64 strips)
    road_min_wmma_kernel<<<grid, 256, 0, stream>>>(hd_map, pred, ndb, drb);

    road_finalize_kernel<<<1, 128, 0, stream>>>(hd_map, pred, ndb, drb, out);
}